// TMQI_SSIM_44753559224321
// MI455X (gfx1250) — compile-verified
//
#include <hip/hip_runtime.h>
#include <stdint.h>

#define TH 22      // output rows per tile
#define TW 64      // output cols per tile
#define SH 32      // TH + 10 (input rows needed)
#define SW 76      // TW + 12 (TW + 10 halo + 2 pad for K=28)
#define OH 1014
#define OW 1014

typedef __attribute__((ext_vector_type(2))) float v2f;
typedef __attribute__((ext_vector_type(8))) float v8f;

// Monotonic float<->uint key so uint atomicMin/Max order == float order.
__device__ __forceinline__ unsigned fkey(float f) {
  unsigned u = __float_as_uint(f);
  return (u & 0x80000000u) ? ~u : (u | 0x80000000u);
}
__device__ __forceinline__ float funkey(unsigned k) {
  unsigned u = (k & 0x80000000u) ? (k & 0x7fffffffu) : ~k;
  return __uint_as_float(u);
}

__global__ void init_ws_k(unsigned* wsu, double* wsd) {
  if (threadIdx.x == 0) {
    wsu[0] = 0xffffffffu; wsu[1] = 0u;   // min1,max1 keys
    wsu[2] = 0xffffffffu; wsu[3] = 0u;   // min2,max2 keys
    wsd[0] = 0.0;                        // s_map sum accumulator
  }
}

__global__ __launch_bounds__(256) void minmax_k(const float4* __restrict__ a,
                                                const float4* __restrict__ b,
                                                unsigned* __restrict__ wsu, int n4) {
  float mn1 = 3.4e38f, mx1 = -3.4e38f, mn2 = 3.4e38f, mx2 = -3.4e38f;
  for (int i = blockIdx.x * blockDim.x + threadIdx.x; i < n4;
       i += gridDim.x * blockDim.x) {
    float4 v = a[i];
    mn1 = fminf(mn1, fminf(fminf(v.x, v.y), fminf(v.z, v.w)));
    mx1 = fmaxf(mx1, fmaxf(fmaxf(v.x, v.y), fmaxf(v.z, v.w)));
    float4 w = b[i];
    mn2 = fminf(mn2, fminf(fminf(w.x, w.y), fminf(w.z, w.w)));
    mx2 = fmaxf(mx2, fmaxf(fmaxf(w.x, w.y), fmaxf(w.z, w.w)));
  }
  __shared__ unsigned bm[4];
  if (threadIdx.x == 0) { bm[0] = 0xffffffffu; bm[1] = 0u; bm[2] = 0xffffffffu; bm[3] = 0u; }
  __syncthreads();
  atomicMin(&bm[0], fkey(mn1));
  atomicMax(&bm[1], fkey(mx1));
  atomicMin(&bm[2], fkey(mn2));
  atomicMax(&bm[3], fkey(mx2));
  __syncthreads();
  if (threadIdx.x == 0) {
    atomicMin(&wsu[0], bm[0]); atomicMax(&wsu[1], bm[1]);
    atomicMin(&wsu[2], bm[2]); atomicMax(&wsu[3], bm[3]);
  }
}

__global__ __launch_bounds__(256) void ssim_k(const float* __restrict__ img1,
                                              const float* __restrict__ img2,
                                              const unsigned* __restrict__ wsu,
                                              double* __restrict__ sum_out) {
  __shared__ float s1[SH * SW];
  __shared__ float s2[SH * SW];
  __shared__ float hch[5][SH * TW];   // horizontal box sums per channel
  __shared__ float red[256];

  const int tid  = threadIdx.x;
  const int lane = tid & 31;
  const int wave = tid >> 5;
  const int ox = blockIdx.x * TW;
  const int oy = blockIdx.y * TH;

  // normalization scalars from pass 1
  const float mn1 = funkey(wsu[0]), mx1 = funkey(wsu[1]);
  const float mn2 = funkey(wsu[2]), mx2 = funkey(wsu[3]);
  const float a1 = 255.0f / (mx1 - mn1 + 1e-6f);
  const float b1 = -mn1 * a1;
  const float a2 = 255.0f / (mx2 - mn2 + 1e-6f);
  const float b2 = -mn2 * a2;

  const float* p1 = img1 + ((size_t)blockIdx.z << 20);
  const float* p2 = img2 + ((size_t)blockIdx.z << 20);

  // ---- async global -> LDS tile load (raw values; clamped at edges) ----
  for (int idx = tid; idx < SH * SW; idx += 256) {
    int r = idx / SW;
    int c = idx - r * SW;
    if (c >= TW + 10) {            // pad columns 74,75: hit zero band rows only
      s1[idx] = 0.0f;
      s2[idx] = 0.0f;
    } else {
      int gy = oy + r; if (gy > 1023) gy = 1023;
      int gx = ox + c; if (gx > 1023) gx = 1023;
      const float* g1 = p1 + (gy << 10) + gx;
      const float* g2 = p2 + (gy << 10) + gx;
      unsigned l1 = (unsigned)(uintptr_t)&s1[idx];
      unsigned l2 = (unsigned)(uintptr_t)&s2[idx];
      asm volatile("global_load_async_to_lds_b32 %0, %1, off"
                   :: "v"(l1), "v"(g1) : "memory");
      asm volatile("global_load_async_to_lds_b32 %0, %1, off"
                   :: "v"(l2), "v"(g2) : "memory");
    }
  }
  asm volatile("s_wait_asynccnt 0x0" ::: "memory");
  __syncthreads();

  // ---- banded box-sum matrix B in registers (same for every tile/channel) ----
  // B[i][n] = 1 iff 0 <= i-n <= 10, i in [0,28) (K padded 26->28), n in [0,16)
  v2f bfrag[7];
  {
    const int n  = lane & 15;
    const int kb = (lane >> 4) << 1;
#pragma unroll
    for (int kc = 0; kc < 7; ++kc) {
#pragma unroll
      for (int v = 0; v < 2; ++v) {
        int i = 4 * kc + kb + v;
        bfrag[kc][v] = (i >= n && i <= n + 10) ? 1.0f : 0.0f;
      }
    }
  }

  // ---- horizontal box sums via chained V_WMMA_F32_16X16X4_F32 ----
  // Each of 8 waves owns one (row-group, col-group) 16x16 sub-tile and
  // accumulates all 5 channels with independent accumulators (no branching:
  // the channel loop is fully unrolled at compile time).
  {
    const int rg = wave >> 2;          // row group 0..1
    const int cg = wave & 3;           // col group 0..3
    const int row = (rg << 4) + (lane & 15);
    const int kb  = (lane >> 4) << 1;

    v8f acc[5] = {v8f{}, v8f{}, v8f{}, v8f{}, v8f{}};
#pragma unroll
    for (int kc = 0; kc < 7; ++kc) {
      // Read the two K-slots once; derive all 5 channel fragments from them.
      float x1v[2], x2v[2];
#pragma unroll
      for (int v = 0; v < 2; ++v) {
        const int col = (cg << 4) + 4 * kc + kb + v;
        const float r1 = s1[row * SW + col];
        const float r2 = s2[row * SW + col];
        x1v[v] = a1 * r1 + b1;
        x2v[v] = a2 * r2 + b2;
      }
      v2f af[5];
      af[0] = v2f{x1v[0], x1v[1]};
      af[1] = v2f{x2v[0], x2v[1]};
      af[2] = v2f{x1v[0] * x1v[0], x1v[1] * x1v[1]};
      af[3] = v2f{x2v[0] * x2v[0], x2v[1] * x2v[1]};
      af[4] = v2f{x1v[0] * x2v[0], x1v[1] * x2v[1]};
#pragma unroll
      for (int ch = 0; ch < 5; ++ch) {
        acc[ch] = __builtin_amdgcn_wmma_f32_16x16x4_f32(
            false, af[ch], false, bfrag[kc], (short)0, acc[ch], false, false);
      }
    }
    const int ocol  = (cg << 4) + (lane & 15);
    const int orow0 = (rg << 4) + ((lane >> 4) << 3);
#pragma unroll
    for (int ch = 0; ch < 5; ++ch) {
#pragma unroll
      for (int v = 0; v < 8; ++v)
        hch[ch][(orow0 + v) * TW + ocol] = acc[ch][v];
    }
  }
  __syncthreads();

  // ---- TMQI constants (U_LDR==U_HDR, SIG_LDR==SIG_HDR) ----
  const float csf  = 100.0f * 2.6f * (0.0192f + 1.824f) * expf(-powf(1.824f, 1.1f));
  const float uhdr = 128.0f / (1.4f * csf);
  const float inv_s_sqrt2 = 3.0f / (uhdr * 1.41421356237f);  // 1/(sig*sqrt(2))
  const float inv121 = 1.0f / 121.0f;

  // ---- vertical 11-sum + per-pixel TMQI math + block reduction ----
  float lsum = 0.0f;
  for (int p = tid; p < TH * TW; p += 256) {
    const int r = p >> 6;   // TW == 64
    const int j = p & 63;
    if (oy + r < OH && ox + j < OW) {
      float S0 = 0.f, S1 = 0.f, S2 = 0.f, S3 = 0.f, S4 = 0.f;
#pragma unroll
      for (int k = 0; k < 11; ++k) {
        const int base = (r + k) * TW + j;
        S0 += hch[0][base];
        S1 += hch[1][base];
        S2 += hch[2][base];
        S3 += hch[3][base];
        S4 += hch[4][base];
      }
      const float mu1 = S0 * inv121, mu2 = S1 * inv121;
      const float var1 = S2 * inv121 - mu1 * mu1;
      const float var2 = S3 * inv121 - mu2 * mu2;
      const float cov  = S4 * inv121 - mu1 * mu2;
      const float sg1 = sqrtf(fmaxf(var1, 0.0f) + 1e-6f);
      const float sg2 = sqrtf(fmaxf(var2, 0.0f) + 1e-6f);
      const float p1c = 0.5f * (1.0f + erff((sg1 - uhdr) * inv_s_sqrt2));
      const float p2c = 0.5f * (1.0f + erff((sg2 - uhdr) * inv_s_sqrt2));
      const float sm = ((2.0f * p1c * p2c + 0.01f) / (p1c * p1c + p2c * p2c + 0.01f))
                     * ((cov + 10.0f) / (sg1 * sg2 + 10.0f));
      lsum += sm;
    }
  }
  red[tid] = lsum;
  __syncthreads();
  for (int s = 128; s > 0; s >>= 1) {
    if (tid < s) red[tid] += red[tid + s];
    __syncthreads();
  }
  if (tid == 0) atomicAdd(sum_out, (double)red[0]);
}

__global__ void finalize_k(const double* __restrict__ sum, float* __restrict__ out) {
  out[0] = (float)(sum[0] / (16.0 * 1014.0 * 1014.0));
}

extern "C" void kernel_launch(void* const* d_in, const int* in_sizes, int n_in,
                              void* d_out, int out_size, void* d_ws, size_t ws_size,
                              hipStream_t stream) {
  const float* img1 = (const float*)d_in[0];
  const float* img2 = (const float*)d_in[1];
  unsigned* wsu = (unsigned*)d_ws;
  double*   wsd = (double*)((char*)d_ws + 16);
  float*    out = (float*)d_out;
  const int n = in_sizes[0];  // 16 * 1024 * 1024

  init_ws_k<<<1, 64, 0, stream>>>(wsu, wsd);
  minmax_k<<<2048, 256, 0, stream>>>((const float4*)img1, (const float4*)img2, wsu, n / 4);

  dim3 grid((OW + TW - 1) / TW, (OH + TH - 1) / TH, 16);
  ssim_k<<<grid, 256, 0, stream>>>(img1, img2, wsu, wsd);

  finalize_k<<<1, 1, 0, stream>>>(wsd, out);
}